// GCBlock_15745350107646
// MI455X (gfx1250) — compile-verified
//
#include <hip/hip_runtime.h>

#define N_NODES 30000
#define N_EDGES 240000
#define C 64
#define NB 16

typedef __attribute__((ext_vector_type(2))) float v2f;
typedef __attribute__((ext_vector_type(8))) float v8f;

static __device__ __forceinline__ v8f wmma4(v2f a, v2f b, v8f c) {
  // V_WMMA_F32_16X16X4_F32 : D = A(16x4) * B(4x16) + C(16x16), fp32 throughout
  return __builtin_amdgcn_wmma_f32_16x16x4_f32(false, a, false, b, (short)0, c,
                                               false, false);
}

// One 16x16 tile: A (row-major, LDS, stride sa), B (row-major global, ld ldb),
// output cols [n0, n0+16).  ln = lane&15, lh = lane>>4 (wave32).
static __device__ __forceinline__ v8f gemm_tile(const float* A, int sa,
                                                const float* __restrict__ B,
                                                int ldb, int n0, int K, int ln,
                                                int lh, v8f acc) {
  for (int k0 = 0; k0 < K; k0 += 4) {
    int ka = k0 + 2 * lh;
    v2f a, b;
    a.x = A[ln * sa + ka];
    a.y = A[ln * sa + ka + 1];
    b.x = B[ka * ldb + n0 + ln];
    b.y = B[(ka + 1) * ldb + n0 + ln];
    acc = wmma4(a, b, acc);
  }
  return acc;
}

// ---------------- node MLP:  Y = relu(X@W1+b1)@W2+b2 ; Yp and Yacc copies ---
__global__ __launch_bounds__(128) void node_mlp_kernel(
    const float* __restrict__ X, const float* __restrict__ W1,
    const float* __restrict__ b1, const float* __restrict__ W2,
    const float* __restrict__ b2, float* __restrict__ Yp,
    float* __restrict__ Yacc) {
  __shared__ float xs[16 * 66];
  __shared__ float hs[16 * 66];
  const int tid = threadIdx.x;
  const int wave = tid >> 5, lane = tid & 31;
  const int ln = lane & 15, lh = lane >> 4;
  const long r0 = (long)blockIdx.x * 16;

  for (int i = tid; i < 16 * C; i += 128) {
    int r = i >> 6, c = i & 63;
    xs[r * 66 + c] = X[(r0 + r) * C + c];
  }
  __syncthreads();
  {
    v8f acc = {};
    const int n0 = wave * 16;
    acc = gemm_tile(xs, 66, W1, C, n0, C, ln, lh, acc);
#pragma unroll
    for (int r = 0; r < 8; ++r) {
      float v = acc[r] + b1[n0 + ln];
      hs[(r + 8 * lh) * 66 + n0 + ln] = v > 0.f ? v : 0.f;
    }
  }
  __syncthreads();
  {
    v8f acc = {};
    const int n0 = wave * 16;
    acc = gemm_tile(hs, 66, W2, C, n0, C, ln, lh, acc);
#pragma unroll
    for (int r = 0; r < 8; ++r) {
      float v = acc[r] + b2[n0 + ln];
      long o = (r0 + r + 8 * lh) * C + n0 + ln;
      Yp[o] = v;
      Yacc[o] = v;
    }
  }
}

// ------------- permute pi1_W2 [64,1024] -> Wp[(h*16+b)*64 + hp] -------------
__global__ void perm_w2_kernel(const float* __restrict__ W2,
                               float* __restrict__ Wp) {
  int d = blockIdx.x * 256 + threadIdx.x;
  if (d >= 64 * NB * 64) return;
  int row = d >> 6, hp = d & 63;
  int h = row >> 4, b = row & 15;
  Wp[d] = W2[h * (64 * NB) + hp * NB + b];
}

// ---------------------------- edge pipeline --------------------------------
__global__ __launch_bounds__(128) void edge_kernel(
    const float* __restrict__ p1p, const float* __restrict__ p3p,
    float* __restrict__ p1a, float* __restrict__ p3a,
    const float* __restrict__ diff, const float* __restrict__ basis,
    const int* __restrict__ idx_i, const int* __restrict__ idx_j,
    const float* __restrict__ piW1, const float* __restrict__ pib1,
    const float* __restrict__ Wp, const float* __restrict__ i1W1,
    const float* __restrict__ i1b1, const float* __restrict__ i1W2,
    const float* __restrict__ i1b2, const float* __restrict__ i3W1,
    const float* __restrict__ i3b1, const float* __restrict__ i3W2,
    const float* __restrict__ i3b2) {
  __shared__ float Xs[16 * 130];  // [p1'[i] | p1'[j]]  [16][128]
  __shared__ float Hs[16 * 66];   // pi1 hidden        [16][64]
  __shared__ float Bs[16 * 18];   // basis             [16][16]
  __shared__ float I1[16 * 66];   // basis-contracted  [16][64]
  __shared__ float Ms[16 * 130];  // ii1 hidden        [16][128]
  __shared__ float Gs[16 * 194];  // gates i1_1|2|3    [16][192]
  __shared__ float X3[48 * 66];   // gated p3 gather   [48][64]
  __shared__ float M3[48 * 130];  // ii3 hidden        [48][128]
  __shared__ int ii[16], jj[16];
  __shared__ float dif[48];

  const int tid = threadIdx.x;
  const int wave = tid >> 5, lane = tid & 31;
  const int ln = lane & 15, lh = lane >> 4;
  const int e0 = blockIdx.x * 16;

  if (tid < 16) { ii[tid] = idx_i[e0 + tid]; jj[tid] = idx_j[e0 + tid]; }
  if (tid < 48) dif[tid] = diff[e0 * 3 + tid];
  for (int i = tid; i < 16 * NB; i += 128)
    Bs[(i >> 4) * 18 + (i & 15)] = basis[e0 * NB + i];
  __syncthreads();

  for (int i = tid; i < 16 * 128; i += 128) {  // gather concat input
    int e = i >> 7, c = i & 127;
    Xs[e * 130 + c] = (c < 64) ? p1p[(long)ii[e] * C + c]
                               : p1p[(long)jj[e] * C + (c - 64)];
  }
  __syncthreads();

  {  // P1: h = X @ pi1_W1 + b1  (no activation, per PILayer)
    v8f acc = {};
    const int n0 = wave * 16;
    acc = gemm_tile(Xs, 130, piW1, C, n0, 128, ln, lh, acc);
#pragma unroll
    for (int r = 0; r < 8; ++r)
      Hs[(r + 8 * lh) * 66 + n0 + ln] = acc[r] + pib1[n0 + ln];
  }
  __syncthreads();

  {  // P2: i1 = (h outer basis) @ Wp  — A fragments built on the fly
    v8f acc = {};
    const int n0 = wave * 16;
#pragma unroll 4
    for (int k0 = 0; k0 < 64 * NB; k0 += 4) {
      int ka = k0 + 2 * lh;  // even => same h-block for ka, ka+1
      float hv = Hs[ln * 66 + (ka >> 4)];
      v2f a, b;
      a.x = hv * Bs[ln * 18 + (ka & 15)];
      a.y = hv * Bs[ln * 18 + ((ka + 1) & 15)];
      b.x = Wp[ka * C + n0 + ln];
      b.y = Wp[(ka + 1) * C + n0 + ln];
      acc = wmma4(a, b, acc);
    }
#pragma unroll
    for (int r = 0; r < 8; ++r) I1[(r + 8 * lh) * 66 + n0 + ln] = acc[r];
  }
  __syncthreads();

  for (int ct = wave; ct < 8; ct += 4) {  // P3: m = relu(i1@ii1_W1+b1)
    v8f acc = {};
    const int n0 = ct * 16;
    acc = gemm_tile(I1, 66, i1W1, 128, n0, 64, ln, lh, acc);
#pragma unroll
    for (int r = 0; r < 8; ++r) {
      float v = acc[r] + i1b1[n0 + ln];
      Ms[(r + 8 * lh) * 130 + n0 + ln] = v > 0.f ? v : 0.f;
    }
  }
  __syncthreads();

  for (int ct = wave; ct < 12; ct += 4) {  // P4: g = m@ii1_W2[:, :192]+b2
    v8f acc = {};
    const int n0 = ct * 16;
    acc = gemm_tile(Ms, 130, i1W2, 4 * C, n0, 128, ln, lh, acc);
#pragma unroll
    for (int r = 0; r < 8; ++r)
      Gs[(r + 8 * lh) * 194 + n0 + ln] = acc[r] + i1b2[n0 + ln];
  }
  __syncthreads();

  // P5: scatter i1_1 into p1 ; build gated gather X3 = p3'[j] * i1_2
  for (int i = tid; i < 16 * C; i += 128) {
    int e = i >> 6, c = i & 63;
    atomicAdd(&p1a[(long)ii[e] * C + c], Gs[e * 194 + c]);
  }
  for (int i = tid; i < 48 * C; i += 128) {
    int row = i >> 6, c = i & 63;
    int e = row / 3, k = row - 3 * e;
    X3[row * 66 + c] =
        p3p[((long)jj[e] * 3 + k) * C + c] * Gs[e * 194 + 64 + c];
  }
  __syncthreads();

  for (int t = wave; t < 24; t += 4) {  // P6: m3 = relu(X3@ii3_W1+b1) [48,128]
    int rt = t >> 3, ct = t & 7;
    v8f acc = {};
    const int n0 = ct * 16;
    acc = gemm_tile(&X3[rt * 16 * 66], 66, i3W1, 128, n0, 64, ln, lh, acc);
#pragma unroll
    for (int r = 0; r < 8; ++r) {
      float v = acc[r] + i3b1[n0 + ln];
      M3[(rt * 16 + r + 8 * lh) * 130 + n0 + ln] = v > 0.f ? v : 0.f;
    }
  }
  __syncthreads();

  for (int t = wave; t < 12; t += 4) {  // P7: y3 + b2 + diff*i1_3, scatter
    int rt = t >> 2, ct = t & 3;
    v8f acc = {};
    const int n0 = ct * 16;
    acc = gemm_tile(&M3[rt * 16 * 130], 130, i3W2, C, n0, 128, ln, lh, acc);
#pragma unroll
    for (int r = 0; r < 8; ++r) {
      int row = rt * 16 + r + 8 * lh;
      int e = row / 3, k = row - 3 * e;
      int c = n0 + ln;
      float v = acc[r] + i3b2[c] + dif[e * 3 + k] * Gs[e * 194 + 128 + c];
      atomicAdd(&p3a[((long)ii[e] * 3 + k) * C + c], v);
    }
  }
}

// ------------------- finalize: DotLayer + ScaleLayer ------------------------
__global__ void finalize_kernel(const float* __restrict__ p1a,
                                const float* __restrict__ p3a,
                                float* __restrict__ out) {
  long i = (long)blockIdx.x * 256 + threadIdx.x;
  if (i >= (long)N_NODES * C) return;
  long n = i >> 6;
  int c = (int)(i & 63);
  float a = p1a[i];
  float s0 = p3a[n * 192 + c];
  float s1 = p3a[n * 192 + 64 + c];
  float s2 = p3a[n * 192 + 128 + c];
  out[i] = a + s0 * s0 + s1 * s1 + s2 * s2;  // next_p1
  float* o3 = out + (long)N_NODES * C;       // next_p3
  o3[n * 192 + c] = s0 * a;
  o3[n * 192 + 64 + c] = s1 * a;
  o3[n * 192 + 128 + c] = s2 * a;
}

extern "C" void kernel_launch(void* const* d_in, const int* in_sizes, int n_in,
                              void* d_out, int out_size, void* d_ws,
                              size_t ws_size, hipStream_t stream) {
  (void)in_sizes; (void)n_in; (void)out_size; (void)ws_size;
  const float* p1 = (const float*)d_in[0];
  const float* p3 = (const float*)d_in[1];
  // d_in[2] (p5), d_in[15..18] (pp5), d_in[30..33] (ii5) are dead code.
  const float* diff = (const float*)d_in[3];
  const float* basis = (const float*)d_in[4];
  const int* idx_i = (const int*)d_in[5];
  const int* idx_j = (const int*)d_in[6];
  const float* pp1_W1 = (const float*)d_in[7];
  const float* pp1_b1 = (const float*)d_in[8];
  const float* pp1_W2 = (const float*)d_in[9];
  const float* pp1_b2 = (const float*)d_in[10];
  const float* pp3_W1 = (const float*)d_in[11];
  const float* pp3_b1 = (const float*)d_in[12];
  const float* pp3_W2 = (const float*)d_in[13];
  const float* pp3_b2 = (const float*)d_in[14];
  const float* pi1_W1 = (const float*)d_in[19];
  const float* pi1_b1 = (const float*)d_in[20];
  const float* pi1_W2 = (const float*)d_in[21];
  const float* ii1_W1 = (const float*)d_in[22];
  const float* ii1_b1 = (const float*)d_in[23];
  const float* ii1_W2 = (const float*)d_in[24];
  const float* ii1_b2 = (const float*)d_in[25];
  const float* ii3_W1 = (const float*)d_in[26];
  const float* ii3_b1 = (const float*)d_in[27];
  const float* ii3_W2 = (const float*)d_in[28];
  const float* ii3_b2 = (const float*)d_in[29];

  float* ws = (float*)d_ws;
  float* p1p = ws;                               // [N,64]  pre-scatter copy
  float* p1a = p1p + (long)N_NODES * C;          // [N,64]  accumulator
  float* p3p = p1a + (long)N_NODES * C;          // [N,3,64]
  float* p3a = p3p + (long)N_NODES * 3 * C;      // [N,3,64]
  float* w2p = p3a + (long)N_NODES * 3 * C;      // [1024,64] permuted pi1_W2

  node_mlp_kernel<<<N_NODES / 16, 128, 0, stream>>>(p1, pp1_W1, pp1_b1, pp1_W2,
                                                    pp1_b2, p1p, p1a);
  node_mlp_kernel<<<(N_NODES * 3) / 16, 128, 0, stream>>>(
      p3, pp3_W1, pp3_b1, pp3_W2, pp3_b2, p3p, p3a);
  perm_w2_kernel<<<(64 * NB * 64 + 255) / 256, 256, 0, stream>>>(pi1_W2, w2p);
  edge_kernel<<<N_EDGES / 16, 128, 0, stream>>>(
      p1p, p3p, p1a, p3a, diff, basis, idx_i, idx_j, pi1_W1, pi1_b1, w2p,
      ii1_W1, ii1_b1, ii1_W2, ii1_b2, ii3_W1, ii3_b1, ii3_W2, ii3_b2);
  finalize_kernel<<<(int)(((long)N_NODES * C + 255) / 256), 256, 0, stream>>>(
      p1a, p3a, (float*)d_out);
}